// DecoupledSTBlock_22076131902168
// MI455X (gfx1250) — compile-verified
//
#include <hip/hip_runtime.h>
#include <hip/hip_bf16.h>
#include <math.h>

// ---------------------------------------------------------------------------
// Problem constants (from reference)
// ---------------------------------------------------------------------------
#define BB 4
#define TT_ 24
#define NN_ 207
#define DD 128
#define DI 256            // D_INNER
#define DS 16             // D_STATE
#define DTR 8             // DT_RANK
#define DCONV 4
#define TOK (BB*TT_*NN_)  // 19872 tokens (multiple of 32)

typedef __attribute__((ext_vector_type(2))) float v2f;
typedef __attribute__((ext_vector_type(8))) float v8f;

// ---------------------------------------------------------------------------
// Generic f32 WMMA GEMM:  C[M,N] = A[M,K] @ W[N,K]^T (+bias) (+act) (+resid)
// Each wave owns a 32x16 C block (two 16x16 accumulators sharing one B
// fragment), K stepped by 4 with V_WMMA_F32_16X16X4_F32. All loads are
// unconditional float2 loads from clamped pointers: OOB columns only affect
// lanes/rows that are never stored, so no predication in the hot loop.
// Requires: M % 32 == 0, K % 4 == 0 (true for every use here).
// perm==1 remaps output rows (b,n,t)->(b,t,n) for the temporal stream.
// ---------------------------------------------------------------------------
__global__ void gemm_wmma_f32(const float* __restrict__ A,
                              const float* __restrict__ W,
                              const float* __restrict__ bias,
                              float* __restrict__ C,
                              const float* __restrict__ resid,
                              int M, int Nn, int K, int act, int perm)
{
    int tilesN  = (Nn + 15) >> 4;
    int tilesM2 = M >> 5;                         // 32-row blocks
    int wave = blockIdx.x * (blockDim.x >> 5) + (threadIdx.x >> 5);
    if (wave >= tilesM2 * tilesN) return;
    int tm = wave / tilesN, tn = wave % tilesN;

    int lane = threadIdx.x & 31;
    int half = lane >> 4;        // 0: K=k0..k0+1   1: K=k0+2..k0+3
    int l16  = lane & 15;

    int row0  = tm * 32 + l16;                    // always < M
    int row1  = row0 + 16;
    int ncol  = tn * 16 + l16;
    int ncolc = (ncol < Nn) ? ncol : (Nn - 1);    // clamped (never stored if OOB)

    const v2f* __restrict__ Ap0 = (const v2f*)(A + (size_t)row0  * K);
    const v2f* __restrict__ Ap1 = (const v2f*)(A + (size_t)row1  * K);
    const v2f* __restrict__ Wp  = (const v2f*)(W + (size_t)ncolc * K);

    v8f acc0 = {0.f,0.f,0.f,0.f,0.f,0.f,0.f,0.f};
    v8f acc1 = {0.f,0.f,0.f,0.f,0.f,0.f,0.f,0.f};

    int kv = half;                                // v2f index = k0/2 + half
#pragma unroll 4
    for (int k0 = 0; k0 < K; k0 += 4, kv += 2) {
        v2f b  = Wp[kv];
        v2f a0 = Ap0[kv];
        v2f a1 = Ap1[kv];
        acc0 = __builtin_amdgcn_wmma_f32_16x16x4_f32(
            false, a0, false, b, (short)0, acc0, false, false);
        acc1 = __builtin_amdgcn_wmma_f32_16x16x4_f32(
            false, a1, false, b, (short)0, acc1, false, false);
    }

    if (ncol >= Nn) return;                       // N-tail columns: discard
    float bv0 = bias ? bias[ncol] : 0.f;

#pragma unroll
    for (int r = 0; r < 8; ++r) {                 // C layout: VGPR r -> rows r / r+8
        int m0 = tm * 32 + half * 8 + r;
        int m1 = m0 + 16;
#pragma unroll
        for (int q = 0; q < 2; ++q) {
            int m = q ? m1 : m0;
            float v = (q ? acc1[r] : acc0[r]) + bv0;
            int orow = m;
            if (perm == 1) {                      // row=(b*NN+n)*TT+t -> (b*TT+t)*NN+n
                int t = m % TT_;
                int rest = m / TT_;
                int n = rest % NN_;
                int b = rest / NN_;
                orow = (b * TT_ + t) * NN_ + n;
            }
            if (act == 1) v = 0.5f * v * (1.f + erff(v * 0.70710678118654752f));
            if (resid) v += resid[(size_t)orow * Nn + ncol];
            C[(size_t)orow * Nn + ncol] = v;
        }
    }
}

// ---------------------------------------------------------------------------
// Batched adjacency GEMM:  AGG[g] = adj(207x207) @ MSG[g](207x128), g=0..95
// K tail handled with clamped-index loads + v_cndmask selects (no exec churn).
// ---------------------------------------------------------------------------
__global__ void agg_gemm_wmma(const float* __restrict__ adj,
                              const float* __restrict__ msg,
                              float* __restrict__ agg)
{
    const int tilesN = DD / 16;                   // 8
    const int tilesM = (NN_ + 15) / 16;           // 13
    int wave = blockIdx.x * (blockDim.x >> 5) + (threadIdx.x >> 5);
    if (wave >= tilesM * tilesN) return;
    int tm = wave / tilesN, tn = wave % tilesN;
    int g = blockIdx.y;

    int lane = threadIdx.x & 31;
    int half = lane >> 4;
    int l16  = lane & 15;

    int arow  = tm * 16 + l16;
    int arowc = (arow < NN_) ? arow : (NN_ - 1);  // clamped A row
    int ncol  = tn * 16 + l16;                    // always < 128

    const float* __restrict__ Ap = adj + (size_t)arowc * NN_;
    const float* __restrict__ Mp = msg + (size_t)g * NN_ * DD + ncol;

    v8f acc = {0.f,0.f,0.f,0.f,0.f,0.f,0.f,0.f};
    // 52 steps of 4 cover K=0..207 (tail zeroed by selects)
    for (int k0 = 0; k0 < 208; k0 += 4) {
        int ka  = k0 + half * 2;
        int k0c = (ka     < NN_) ? ka     : (NN_ - 1);
        int k1c = (ka + 1 < NN_) ? ka + 1 : (NN_ - 1);
        float a0 = Ap[k0c], a1 = Ap[k1c];
        float b0 = Mp[(size_t)k0c * DD], b1 = Mp[(size_t)k1c * DD];
        v2f av, bv;
        av.x = (ka     < NN_) ? a0 : 0.f;
        av.y = (ka + 1 < NN_) ? a1 : 0.f;
        bv.x = (ka     < NN_) ? b0 : 0.f;
        bv.y = (ka + 1 < NN_) ? b1 : 0.f;
        acc = __builtin_amdgcn_wmma_f32_16x16x4_f32(
            false, av, false, bv, (short)0, acc, false, false);
    }
#pragma unroll
    for (int r = 0; r < 8; ++r) {
        int m = tm * 16 + half * 8 + r;
        if (m < NN_) agg[((size_t)g * NN_ + m) * DD + ncol] = acc[r];
    }
}

// ---------------------------------------------------------------------------
// Elementwise / per-token kernels
// ---------------------------------------------------------------------------
__global__ void transpose_btnd_bntd(const float* __restrict__ x, float* __restrict__ o)
{
    int idx = blockIdx.x * blockDim.x + threadIdx.x;
    if (idx >= TOK * DD) return;
    int d = idx % DD;
    int t = (idx / DD) % TT_;
    int n = (idx / (DD * TT_)) % NN_;
    int b =  idx / (DD * TT_ * NN_);
    o[idx] = x[(((size_t)b * TT_ + t) * NN_ + n) * DD + d];
}

__global__ void ln_kernel(const float* __restrict__ in, float* __restrict__ out,
                          const float* __restrict__ g, const float* __restrict__ b)
{
    __shared__ float red[DD];
    int tok = blockIdx.x, d = threadIdx.x;
    float v = in[(size_t)tok * DD + d];
    red[d] = v; __syncthreads();
    for (int s = 64; s > 0; s >>= 1) { if (d < s) red[d] += red[d + s]; __syncthreads(); }
    float mu = red[0] * (1.f / DD); __syncthreads();
    float c = v - mu;
    red[d] = c * c; __syncthreads();
    for (int s = 64; s > 0; s >>= 1) { if (d < s) red[d] += red[d + s]; __syncthreads(); }
    float var = red[0] * (1.f / DD);
    out[(size_t)tok * DD + d] = c * rsqrtf(var + 1e-5f) * g[d] + b[d];
}

__global__ void conv_silu_kernel(const float* __restrict__ XZ,
                                 const float* __restrict__ cw,
                                 const float* __restrict__ cb,
                                 float* __restrict__ U, int NSEQ, int L)
{
    int idx = blockIdx.x * blockDim.x + threadIdx.x;
    if (idx >= NSEQ * L * DI) return;
    int c = idx % DI;
    int t = (idx / DI) % L;
    int s =  idx / (DI * L);
    float acc = cb[c];
#pragma unroll
    for (int j = 0; j < DCONV; ++j) {
        int tt = t - (DCONV - 1) + j;
        int ttc = (tt >= 0) ? tt : 0;                              // clamped load
        float xv = XZ[((size_t)s * L + ttc) * (2 * DI) + c];
        acc += ((tt >= 0) ? cw[c * DCONV + j] : 0.f) * xv;         // cndmask, not branch
    }
    U[((size_t)s * L + t) * DI + c] = acc / (1.f + __expf(-acc));  // silu
}

__global__ void dt_kernel(const float* __restrict__ DBL,
                          const float* __restrict__ dtw,
                          const float* __restrict__ dtb,
                          float* __restrict__ DT)
{
    int idx = blockIdx.x * blockDim.x + threadIdx.x;
    if (idx >= TOK * DI) return;
    int d = idx % DI;
    size_t tok = idx / DI;
    float acc = dtb[d];
#pragma unroll
    for (int r = 0; r < DTR; ++r) acc += DBL[tok * 40 + r] * dtw[d * DTR + r];
    DT[idx] = (acc > 20.f) ? acc : log1pf(__expf(acc));            // softplus
}

// selective scan: one block per sequence, one thread per d_inner channel
__global__ void scan_kernel(const float* __restrict__ DT,
                            const float* __restrict__ DBL,
                            const float* __restrict__ U,
                            const float* __restrict__ XZ,
                            const float* __restrict__ A_log,
                            const float* __restrict__ Dv,
                            float* __restrict__ Y, int L)
{
    int d = threadIdx.x;
    int s = blockIdx.x;
    float Aa[DS], h[DS];
#pragma unroll
    for (int i = 0; i < DS; ++i) { Aa[i] = -__expf(A_log[d * DS + i]); h[i] = 0.f; }
    float Dp = Dv[d];
    for (int t = 0; t < L; ++t) {
        size_t tok = (size_t)s * L + t;
        float dt = DT[tok * DI + d];
        float u  = U [tok * DI + d];
        const float* Bp = DBL + tok * 40 + DTR;
        const float* Cp = DBL + tok * 40 + DTR + DS;
        float dtu = dt * u, acc = 0.f;
#pragma unroll
        for (int i = 0; i < DS; ++i) {
            h[i] = __expf(dt * Aa[i]) * h[i] + dtu * Bp[i];
            acc += h[i] * Cp[i];
        }
        float z = XZ[tok * (2 * DI) + DI + d];
        Y[tok * DI + d] = (acc + u * Dp) * (z / (1.f + __expf(-z)));
    }
}

__global__ void concat_kernel(const float* __restrict__ A, const float* __restrict__ Bb,
                              float* __restrict__ O)
{
    int idx = blockIdx.x * blockDim.x + threadIdx.x;
    if (idx >= TOK * 2 * DD) return;
    int col = idx % (2 * DD);
    size_t tok = idx / (2 * DD);
    const float* src = (col < DD) ? (A + tok * DD + col) : (Bb + tok * DD + (col - DD));
    O[idx] = *src;
}

// gates = softmax(h@w2^T+b2); fused = g0*xt+g1*xsp; out = LN(fused)
__global__ void fusion_final_kernel(const float* __restrict__ H,
                                    const float* __restrict__ XT,
                                    const float* __restrict__ XSP,
                                    const float* __restrict__ w2,
                                    const float* __restrict__ b2,
                                    const float* __restrict__ g,
                                    const float* __restrict__ bb,
                                    float* __restrict__ out)
{
    __shared__ float r0[DD], r1[DD];
    int tok = blockIdx.x, d = threadIdx.x;
    float h = H[(size_t)tok * DD + d];
    r0[d] = h * w2[d];
    r1[d] = h * w2[DD + d];
    __syncthreads();
    for (int s = 64; s > 0; s >>= 1) {
        if (d < s) { r0[d] += r0[d + s]; r1[d] += r1[d + s]; }
        __syncthreads();
    }
    float l0 = r0[0] + b2[0], l1 = r1[0] + b2[1];
    __syncthreads();
    float mx = fmaxf(l0, l1);
    float e0 = __expf(l0 - mx), e1 = __expf(l1 - mx);
    float inv = 1.f / (e0 + e1);
    float f = (e0 * inv) * XT[(size_t)tok * DD + d] + (e1 * inv) * XSP[(size_t)tok * DD + d];
    r0[d] = f; __syncthreads();
    for (int s = 64; s > 0; s >>= 1) { if (d < s) r0[d] += r0[d + s]; __syncthreads(); }
    float mu = r0[0] * (1.f / DD); __syncthreads();
    float c = f - mu;
    r0[d] = c * c; __syncthreads();
    for (int s = 64; s > 0; s >>= 1) { if (d < s) r0[d] += r0[d + s]; __syncthreads(); }
    float var = r0[0] * (1.f / DD);
    out[(size_t)tok * DD + d] = c * rsqrtf(var + 1e-5f) * g[d] + bb[d];
}

// ---------------------------------------------------------------------------
// Host-side orchestration
// ---------------------------------------------------------------------------
static inline void launch_gemm(const float* A, const float* W, const float* bias,
                               float* C, const float* resid,
                               int M, int N, int K, int act, int perm, hipStream_t s)
{
    int tiles = (M / 32) * ((N + 15) / 16);
    int blocks = (tiles + 7) / 8;                 // 8 waves (wave32) per 256-thread block
    gemm_wmma_f32<<<blocks, 256, 0, s>>>(A, W, bias, C, resid, M, N, K, act, perm);
}

struct MambaP {
    const float *in_proj_w, *conv_w, *conv_b, *x_proj_w, *dt_proj_w, *dt_proj_b;
    const float *A_log, *Dv, *out_proj_w, *norm_g, *norm_b;
};

static void run_mamba(const MambaP& p,
                      const float* seq_in,                 // tokens x 128, sequence order
                      float* BLN, float* XZ, float* U, float* DBL, float* DT, float* Y,
                      float* OUT, const float* resid, int perm,
                      int NSEQ, int L, hipStream_t s)
{
    ln_kernel<<<TOK, DD, 0, s>>>(seq_in, BLN, p.norm_g, p.norm_b);
    launch_gemm(BLN, p.in_proj_w, nullptr, XZ, nullptr, TOK, 2 * DI, DD, 0, 0, s);
    {
        int total = NSEQ * L * DI;
        conv_silu_kernel<<<(total + 255) / 256, 256, 0, s>>>(XZ, p.conv_w, p.conv_b, U, NSEQ, L);
    }
    launch_gemm(U, p.x_proj_w, nullptr, DBL, nullptr, TOK, DTR + 2 * DS, DI, 0, 0, s);
    {
        int total = TOK * DI;
        dt_kernel<<<(total + 255) / 256, 256, 0, s>>>(DBL, p.dt_proj_w, p.dt_proj_b, DT);
    }
    scan_kernel<<<NSEQ, DI, 0, s>>>(DT, DBL, U, XZ, p.A_log, p.Dv, Y, L);
    launch_gemm(Y, p.out_proj_w, nullptr, OUT, resid, TOK, DD, DI, 0, perm, s);
}

static MambaP make_mamba(void* const* d_in, int base)
{
    MambaP p;
    p.in_proj_w = (const float*)d_in[base + 0];
    p.conv_w    = (const float*)d_in[base + 1];
    p.conv_b    = (const float*)d_in[base + 2];
    p.x_proj_w  = (const float*)d_in[base + 3];
    p.dt_proj_w = (const float*)d_in[base + 4];
    p.dt_proj_b = (const float*)d_in[base + 5];
    p.A_log     = (const float*)d_in[base + 6];
    p.Dv        = (const float*)d_in[base + 7];
    p.out_proj_w= (const float*)d_in[base + 8];
    p.norm_g    = (const float*)d_in[base + 9];
    p.norm_b    = (const float*)d_in[base + 10];
    return p;
}

extern "C" void kernel_launch(void* const* d_in, const int* in_sizes, int n_in,
                              void* d_out, int out_size, void* d_ws, size_t ws_size,
                              hipStream_t stream)
{
    (void)in_sizes; (void)n_in; (void)out_size; (void)ws_size;

    const float* x   = (const float*)d_in[0];
    const float* adj = (const float*)d_in[1];
    MambaP tm = make_mamba(d_in, 2);
    MambaP sm = make_mamba(d_in, 13);
    const float* msg_w = (const float*)d_in[24];
    const float* msg_b = (const float*)d_in[25];
    const float* upd_w = (const float*)d_in[26];
    const float* upd_b = (const float*)d_in[27];
    const float* gnm_g = (const float*)d_in[28];
    const float* gnm_b = (const float*)d_in[29];
    const float* f_w1  = (const float*)d_in[30];
    const float* f_b1  = (const float*)d_in[31];
    const float* f_w2  = (const float*)d_in[32];
    const float* f_b2  = (const float*)d_in[33];
    const float* f_ng  = (const float*)d_in[34];
    const float* f_nb  = (const float*)d_in[35];
    float* out = (float*)d_out;

    // workspace layout (floats)
    float* ws  = (float*)d_ws;
    float* A_   = ws;                       // TOK*128  seq-ordered mamba input / residual
    float* BLN  = A_   + (size_t)TOK * 128; // TOK*128
    float* XZ   = BLN  + (size_t)TOK * 128; // TOK*512
    float* U    = XZ   + (size_t)TOK * 512; // TOK*256
    float* DBL  = U    + (size_t)TOK * 256; // TOK*40
    float* DT   = DBL  + (size_t)TOK * 40;  // TOK*256
    float* Y    = DT   + (size_t)TOK * 256; // TOK*256
    float* XT   = Y    + (size_t)TOK * 256; // TOK*128  temporal stream result (b,t,n)
    float* XSP  = XT   + (size_t)TOK * 128; // TOK*128  spatial stream result (b,t,n)

    // ---------------- temporal stream: mamba over T per (b,n) ----------------
    {
        int total = TOK * DD;
        transpose_btnd_bntd<<<(total + 255) / 256, 256, 0, stream>>>(x, A_);
    }
    // out rows permuted back to (b,t,n); residual = original x
    run_mamba(tm, A_, BLN, XZ, U, DBL, DT, Y, XT, x, /*perm=*/1,
              /*NSEQ=*/BB * NN_, /*L=*/TT_, stream);

    // ---------------- spatial stream: graph MP + mamba over N ----------------
    launch_gemm(x, msg_w, msg_b, BLN, nullptr, TOK, DD, DD, 0, 0, stream);   // msg
    {
        const int tilesM = (NN_ + 15) / 16, tilesN = DD / 16;
        dim3 grid((tilesM * tilesN + 7) / 8, BB * TT_);
        agg_gemm_wmma<<<grid, 256, 0, stream>>>(adj, BLN, Y);                // agg -> Y
    }
    {
        int total = TOK * 2 * DD;
        concat_kernel<<<(total + 255) / 256, 256, 0, stream>>>(x, Y, U);     // [x, agg]
    }
    launch_gemm(U, upd_w, upd_b, DT, x, TOK, DD, 2 * DD, 0, 0, stream);      // upd + x
    ln_kernel<<<TOK, DD, 0, stream>>>(DT, A_, gnm_g, gnm_b);                 // xsp_in
    run_mamba(sm, A_, BLN, XZ, U, DBL, DT, Y, XSP, A_, /*perm=*/0,
              /*NSEQ=*/BB * TT_, /*L=*/NN_, stream);

    // ---------------- gated dual-stream fusion ----------------
    {
        int total = TOK * 2 * DD;
        concat_kernel<<<(total + 255) / 256, 256, 0, stream>>>(XT, XSP, U);  // [xt, xsp]
    }
    launch_gemm(U, f_w1, f_b1, DT, nullptr, TOK, DD, 2 * DD, /*gelu*/1, 0, stream);
    fusion_final_kernel<<<TOK, DD, 0, stream>>>(DT, XT, XSP, f_w2, f_b2, f_ng, f_nb, out);
}